// Rank_loss_30666066494150
// MI455X (gfx1250) — compile-verified
//
#include <hip/hip_runtime.h>
#include <hip/hip_bf16.h>

// ---------------------------------------------------------------------------
// Rank loss (fused euclidean-dist + masked AP/AN + hardest-pos/neg) for
// MI455X (gfx1250).  Gram matrix via V_WMMA_F32_16X16X4_F32 (full f32
// precision).  B tiles are staged into LDS (double-buffered, via
// GLOBAL_LOAD_ASYNC_TO_LDS_B128 when available) so the 8192x8192 distance
// matrix never touches HBM and WMMAs are not load-latency serialized.
// ---------------------------------------------------------------------------

typedef __attribute__((ext_vector_type(2))) float v2f;
typedef __attribute__((ext_vector_type(8))) float v8f;
typedef __attribute__((ext_vector_type(4))) int   v4i;

constexpr int N = 8192;       // rows
constexpr int D = 256;        // feature dim
constexpr int S = 8;          // column chunks (parallel split of j-loop)
constexpr int KSTEPS = D / 4; // 64 wmma k-steps of K=4
constexpr int COLS_PER_S = N / S;          // 1024
constexpr int JTILES = COLS_PER_S / 16;    // 64 column tiles per chunk
constexpr int ROWS_PER_BLOCK = 128;        // 8 waves x 16 rows
constexpr int ROWBLOCKS = N / ROWS_PER_BLOCK; // 64
constexpr int NFIELD = 10;
constexpr int BSTRIDE = 260;  // padded LDS row stride (floats): 260 % 64 == 4
                              // -> ds_load_b64 B-fragment reads hit all 64
                              // banks exactly once across the wave.

#define MARGIN_1 1.0f
#define MARGIN_2 1.5f
#define ALPHA_1  2.4f
#define ALPHA_2  2.2f
#define EPSV     1e-5f

#if __has_builtin(__builtin_amdgcn_global_load_async_to_lds_b128)
#define HAVE_ASYNC_LDS 1
typedef __attribute__((address_space(1))) v4i gv4i;  // global int4
typedef __attribute__((address_space(3))) v4i lv4i;  // LDS int4
#else
#define HAVE_ASYNC_LDS 0
#endif

__device__ __forceinline__ float wsum16(float v) {
  v += __shfl_xor(v, 1, 32);
  v += __shfl_xor(v, 2, 32);
  v += __shfl_xor(v, 4, 32);
  v += __shfl_xor(v, 8, 32);
  return v;
}
__device__ __forceinline__ float wmax16(float v) {
  v = fmaxf(v, __shfl_xor(v, 1, 32));
  v = fmaxf(v, __shfl_xor(v, 2, 32));
  v = fmaxf(v, __shfl_xor(v, 4, 32));
  v = fmaxf(v, __shfl_xor(v, 8, 32));
  return v;
}
__device__ __forceinline__ float wmin16(float v) {
  v = fminf(v, __shfl_xor(v, 1, 32));
  v = fminf(v, __shfl_xor(v, 2, 32));
  v = fminf(v, __shfl_xor(v, 4, 32));
  v = fminf(v, __shfl_xor(v, 8, 32));
  return v;
}

// ---------------------------------------------------------------------------
// Cooperative staging of one 16x256 f32 column tile into LDS (padded rows).
// 256 threads x 4 chunks of 16B each = 16 KB.
// ---------------------------------------------------------------------------
#if HAVE_ASYNC_LDS
__device__ __forceinline__ void stage_async(const float* __restrict__ x,
                                            int colBase, float* buf, int t) {
#pragma unroll
  for (int q = 0; q < 4; ++q) {
    const int f = q * 256 + t;          // float4 chunk index 0..1023
    const int row = f >> 6;             // 0..15
    const int c4 = f & 63;              // 0..63
    const float* g = x + (size_t)(colBase + row) * D + c4 * 4;
    float* l = buf + row * BSTRIDE + c4 * 4;
    __builtin_amdgcn_global_load_async_to_lds_b128(
        (gv4i*)(uintptr_t)g,
        (lv4i*)(uint32_t)(uintptr_t)l,
        0, 0);
  }
}
__device__ __forceinline__ void wait_async() {
#if __has_builtin(__builtin_amdgcn_s_wait_asynccnt)
  __builtin_amdgcn_s_wait_asynccnt(0);
#else
  asm volatile("s_wait_asynccnt 0x0" ::: "memory");
#endif
}
#else
__device__ __forceinline__ void stage_load(const float* __restrict__ x,
                                           int colBase, float4* stg, int t) {
#pragma unroll
  for (int q = 0; q < 4; ++q) {
    const int f = q * 256 + t;
    const int row = f >> 6;
    const int c4 = f & 63;
    stg[q] = *(const float4*)(x + (size_t)(colBase + row) * D + c4 * 4);
  }
}
__device__ __forceinline__ void stage_store(float* buf, const float4* stg, int t) {
#pragma unroll
  for (int q = 0; q < 4; ++q) {
    const int f = q * 256 + t;
    const int row = f >> 6;
    const int c4 = f & 63;
    *(float4*)(buf + row * BSTRIDE + c4 * 4) = stg[q];
  }
}
#endif

// ---------------------------------------------------------------------------
// Kernel 1: per-row squared norms  sq[i] = sum_k x[i][k]^2
// ---------------------------------------------------------------------------
__global__ void rl_sq_kernel(const float* __restrict__ x, float* __restrict__ sq) {
  __shared__ float red[256];
  const int row = blockIdx.x;
  const int t = threadIdx.x;
  float v = x[(size_t)row * D + t];
  red[t] = v * v;
  __syncthreads();
  for (int s = 128; s > 0; s >>= 1) {
    if (t < s) red[t] += red[t + s];
    __syncthreads();
  }
  if (t == 0) sq[row] = red[0];
}

// ---------------------------------------------------------------------------
// Kernel 2: fused Gram(WMMA f32) + distance + masked accumulators.
// Grid: ROWBLOCKS * S blocks of 256 threads (8 waves).  Wave w of block
// (rb, s) owns rows [rb*128 + w*16, +16) against columns [s*1024, +1024).
// Partial results per (s,row): 10 floats
//   {ap_intra_sum, ap_intra_cnt, ap_cross_sum, ap_cross_cnt,
//    an_intra_num, an_intra_den, an_cross_num, an_cross_den, dap, dan}
// ---------------------------------------------------------------------------
__global__ __launch_bounds__(256, 1)
void rl_main_kernel(const float* __restrict__ x,
                    const int* __restrict__ tgt,
                    const float* __restrict__ sq,
                    float* __restrict__ partial) {
  __shared__ float ldsb[2][16 * BSTRIDE];   // double-buffered B tile, 33 KB

  const int t = threadIdx.x;
  const int lane = t & 31;
  const int wave = t >> 5;
  const int rb = blockIdx.x & (ROWBLOCKS - 1);
  const int s  = blockIdx.x >> 6;                 // ROWBLOCKS == 64

  const int rowBase  = rb * ROWS_PER_BLOCK + wave * 16;
  const int colChunk = s * COLS_PER_S;

  // Modality split at N/2=4096 is a multiple of chunk and tile sizes, so
  // intra/cross is uniform for this wave.
  const bool intra  = ((rowBase >= N / 2) == (colChunk >= N / 2));
  const float margin = intra ? MARGIN_1 : MARGIN_2;
  const float alpha  = intra ? ALPHA_1  : ALPHA_2;

  const int l15 = lane & 15;
  const int hi  = lane >> 4;

  // ---- A fragments for this wave's 16x256 row tile, kept in VGPRs.
  // A 16x4 f32 layout: lanes 0-15 row M=lane hold K={k0,k0+1}; lanes 16-31
  // hold K={k0+2,k0+3} of row M=lane-16.
  const v2f* xa = (const v2f*)(x + (size_t)(rowBase + l15) * D + 2 * hi);
  v2f afrag[KSTEPS];
#pragma unroll
  for (int kk = 0; kk < KSTEPS; ++kk) afrag[kk] = xa[kk * 2];

  // C/D layout: VGPR r, lanes 0-15 -> (M=r, N=lane); lanes 16-31 -> (M=r+8).
  const int rowTop = rowBase + 8 * hi;
  float sqi[8];
  int ti[8];
#pragma unroll
  for (int r = 0; r < 8; ++r) {
    sqi[r] = sq[rowTop + r];
    ti[r]  = tgt[rowTop + r];
  }

  float aps[8], apc[8], ann[8], awd[8], dap[8], dan[8];
#pragma unroll
  for (int r = 0; r < 8; ++r) {
    aps[r] = 0.0f; apc[r] = 0.0f; ann[r] = 0.0f; awd[r] = 0.0f;
    dap[r] = -__builtin_inff();
    dan[r] =  __builtin_inff();
  }

  // ---- Prologue: stage tile 0 into buffer 0.
#if HAVE_ASYNC_LDS
  stage_async(x, colChunk, &ldsb[0][0], t);
  wait_async();
  __syncthreads();
#else
  {
    float4 stg[4];
    stage_load(x, colChunk, stg, t);
    stage_store(&ldsb[0][0], stg, t);
  }
  __syncthreads();
#endif

  for (int jt = 0; jt < JTILES; ++jt) {
    const int p = jt & 1;
    const int colBase = colChunk + jt * 16;
    const int jcol = colBase + l15;
    const float sqj = sq[jcol];
    const int tj = tgt[jcol];

    // Kick off the copy of the next tile into the other buffer while we
    // compute on this one.
#if HAVE_ASYNC_LDS
    if (jt + 1 < JTILES) stage_async(x, colBase + 16, &ldsb[p ^ 1][0], t);
#else
    float4 stg[4];
    if (jt + 1 < JTILES) stage_load(x, colBase + 16, stg, t);
#endif

    // B 4x16 layout (B = x^T): lanes 0-15 col N=lane hold K={k0,k0+1};
    // lanes 16-31 hold K={k0+2,k0+3}.  Read from the padded LDS tile:
    // dword offset = l15*260 + kk*4 + 2*hi  -> all 64 banks touched once.
    const v2f* pb = (const v2f*)&ldsb[p][l15 * BSTRIDE + 2 * hi];

    // 4 independent accumulator chains for WMMA ILP.
    v8f cc[4];
#pragma unroll
    for (int u = 0; u < 4; ++u) cc[u] = (v8f){};

#pragma unroll
    for (int kg = 0; kg < KSTEPS; kg += 8) {
      v2f bf[8];
#pragma unroll
      for (int u = 0; u < 8; ++u) bf[u] = pb[(kg + u) * 2];
#pragma unroll
      for (int u = 0; u < 8; ++u) {
        const int kk = kg + u;
        cc[kk & 3] = __builtin_amdgcn_wmma_f32_16x16x4_f32(
            false, afrag[kk], false, bf[u], (short)0, cc[kk & 3], false, false);
      }
    }

#pragma unroll
    for (int r = 0; r < 8; ++r) {
      float dot = (cc[0][r] + cc[1][r]) + (cc[2][r] + cc[3][r]);
      float d2 = sqi[r] + sqj - 2.0f * dot;
      float d = sqrtf(fmaxf(d2, 1e-12f));
      bool pos = (ti[r] == tj);
      bool diag = ((rowTop + r) == jcol);
      if (pos) {
        if (!diag) {
          aps[r] += fmaxf(d + (margin - alpha), 0.0f);
          apc[r] += 1.0f;
        }
        dap[r] = fmaxf(dap[r], d);           // hardest positive (incl. diag)
      } else {
        if (d < alpha) {
          float diff = alpha - d;
          float w = expf(diff);              // TVAL == 1
          ann[r] += w * diff;
          awd[r] += w;
        }
        dan[r] = fminf(dan[r], d);           // hardest negative
      }
    }

    if (jt + 1 < JTILES) {
#if HAVE_ASYNC_LDS
      wait_async();
#else
      stage_store(&ldsb[p ^ 1][0], stg, t);
#endif
      __syncthreads();
    }
  }

  // Reduce across the 16 lanes sharing each row half.
#pragma unroll
  for (int r = 0; r < 8; ++r) {
    aps[r] = wsum16(aps[r]);
    apc[r] = wsum16(apc[r]);
    ann[r] = wsum16(ann[r]);
    awd[r] = wsum16(awd[r]);
    dap[r] = wmax16(dap[r]);
    dan[r] = wmin16(dan[r]);
  }

  if (l15 == 0) {  // lanes 0 and 16 cover rows rowTop..rowTop+7 of each half
#pragma unroll
    for (int r = 0; r < 8; ++r) {
      float* p2 = partial + ((size_t)s * N + rowTop + r) * NFIELD;
      if (intra) {
        p2[0] = aps[r]; p2[1] = apc[r]; p2[2] = 0.0f; p2[3] = 0.0f;
        p2[4] = ann[r]; p2[5] = awd[r]; p2[6] = 0.0f; p2[7] = 0.0f;
      } else {
        p2[0] = 0.0f; p2[1] = 0.0f; p2[2] = aps[r]; p2[3] = apc[r];
        p2[4] = 0.0f; p2[5] = 0.0f; p2[6] = ann[r]; p2[7] = awd[r];
      }
      p2[8] = dap[r];
      p2[9] = dan[r];
    }
  }
}

// ---------------------------------------------------------------------------
// Kernel 3: combine S partials per row -> per-row loss and correct flag.
// ---------------------------------------------------------------------------
__global__ void rl_combine_kernel(const float* __restrict__ partial,
                                  float* __restrict__ rowloss,
                                  float* __restrict__ rowcorr) {
  const int row = blockIdx.x * blockDim.x + threadIdx.x;
  if (row >= N) return;
  float f[8] = {0, 0, 0, 0, 0, 0, 0, 0};
  float dap = -__builtin_inff();
  float dan =  __builtin_inff();
  for (int s = 0; s < S; ++s) {
    const float* p = partial + ((size_t)s * N + row) * NFIELD;
#pragma unroll
    for (int k = 0; k < 8; ++k) f[k] += p[k];
    dap = fmaxf(dap, p[8]);
    dan = fminf(dan, p[9]);
  }
  float loss = f[0] / (f[1] + EPSV) + f[2] / (f[3] + EPSV)
             + f[4] / (f[5] + EPSV) + f[6] / (f[7] + EPSV);
  rowloss[row] = loss;
  rowcorr[row] = (dan >= dap) ? 1.0f : 0.0f;
}

// ---------------------------------------------------------------------------
// Kernel 4: deterministic final reduction -> out[0]=mean loss, out[1]=correct
// ---------------------------------------------------------------------------
__global__ void rl_finalize_kernel(const float* __restrict__ rowloss,
                                   const float* __restrict__ rowcorr,
                                   float* __restrict__ out) {
  __shared__ float sl[256];
  __shared__ float sc[256];
  const int t = threadIdx.x;
  float a = 0.0f, b = 0.0f;
  for (int i = t; i < N; i += 256) {
    a += rowloss[i];
    b += rowcorr[i];
  }
  sl[t] = a; sc[t] = b;
  __syncthreads();
  for (int s = 128; s > 0; s >>= 1) {
    if (t < s) { sl[t] += sl[t + s]; sc[t] += sc[t + s]; }
    __syncthreads();
  }
  if (t == 0) {
    out[0] = sl[0] / (float)N;
    out[1] = sc[0];
  }
}

// ---------------------------------------------------------------------------
extern "C" void kernel_launch(void* const* d_in, const int* in_sizes, int n_in,
                              void* d_out, int out_size, void* d_ws, size_t ws_size,
                              hipStream_t stream) {
  const float* x   = (const float*)d_in[0];
  const int*   tgt = (const int*)d_in[1];
  float* out = (float*)d_out;

  float* ws_sq      = (float*)d_ws;
  float* ws_partial = ws_sq + N;
  float* ws_rowloss = ws_partial + (size_t)S * N * NFIELD;
  float* ws_rowcorr = ws_rowloss + N;

  rl_sq_kernel<<<N, 256, 0, stream>>>(x, ws_sq);
  rl_main_kernel<<<ROWBLOCKS * S, 256, 0, stream>>>(x, tgt, ws_sq, ws_partial);
  rl_combine_kernel<<<N / 256, 256, 0, stream>>>(ws_partial, ws_rowloss, ws_rowcorr);
  rl_finalize_kernel<<<1, 256, 0, stream>>>(ws_rowloss, ws_rowcorr, out);
}